// ConstrainedDecoderForSingleDec_7421703487622
// MI455X (gfx1250) — compile-verified
//
#include <hip/hip_runtime.h>
#include <hip/hip_bf16.h>
#include <math.h>

typedef __attribute__((ext_vector_type(2))) float v2f;
typedef __attribute__((ext_vector_type(8))) float v8f;

// ---------------------------------------------------------------------------
// Generic WMMA f32 GEMM:  C[M,N] = act( sum_p A_p @ W_p^T  + bias0 + bias1 )
//   A_p : [M, K_p] row-major (lda == K_p)
//   W_p : row-major with leading dim ldw_p; logical B matrix B[k][n] = W_p[n][k]
// Each wave computes a 16x64 output slab: one A fragment feeds 4 WMMAs with
// independent accumulators (4x less A traffic, 4 independent dep chains).
// N must be a multiple of 64; slab count must be a multiple of 8
// (blockDim = 256 = 8 waves). Guard is wave-uniform so EXEC is all-ones for
// every V_WMMA_F32_16X16X4_F32.
// ---------------------------------------------------------------------------
__global__ __launch_bounds__(256) void wmma_gemm3(
    const float* __restrict__ A0, const float* __restrict__ W0, int K0, int ldw0,
    const float* __restrict__ A1, const float* __restrict__ W1, int K1, int ldw1,
    const float* __restrict__ A2, const float* __restrict__ W2, int K2, int ldw2,
    const float* __restrict__ bias0, const float* __restrict__ bias1,
    float* __restrict__ C, int M, int N, int ldc, int col_off, int act)
{
    const int wid  = threadIdx.x >> 5;
    const int lane = threadIdx.x & 31;
    const int slab = blockIdx.x * 8 + wid;        // 16x64 output slab index
    const int nSlabs = N >> 6;
    const int tm = slab / nSlabs;
    const int tn = slab - tm * nSlabs;
    if (tm >= (M >> 4)) return;                   // wave-uniform exit

    const int row  = lane & 15;                   // M index (A) / N index (B,C)
    const int half = lane >> 4;                   // selects K sub-pair
    const int m0 = tm << 4, n0 = tn << 6;

    v8f acc0 = {}, acc1 = {}, acc2 = {}, acc3 = {};
    {
        const float* a = A0 + (long)(m0 + row) * K0 + 2 * half;
        const float* w = W0 + (long)(n0 + row) * ldw0 + 2 * half;
        const long wstep = (long)16 * ldw0;
        for (int k = 0; k < K0; k += 4) {
            v2f av = *(const v2f*)(a + k);
            v2f b0 = *(const v2f*)(w + k);
            v2f b1 = *(const v2f*)(w + wstep + k);
            v2f b2 = *(const v2f*)(w + 2 * wstep + k);
            v2f b3 = *(const v2f*)(w + 3 * wstep + k);
            acc0 = __builtin_amdgcn_wmma_f32_16x16x4_f32(false, av, false, b0, (short)0, acc0, false, false);
            acc1 = __builtin_amdgcn_wmma_f32_16x16x4_f32(false, av, false, b1, (short)0, acc1, false, false);
            acc2 = __builtin_amdgcn_wmma_f32_16x16x4_f32(false, av, false, b2, (short)0, acc2, false, false);
            acc3 = __builtin_amdgcn_wmma_f32_16x16x4_f32(false, av, false, b3, (short)0, acc3, false, false);
        }
    }
    if (A1) {
        const float* a = A1 + (long)(m0 + row) * K1 + 2 * half;
        const float* w = W1 + (long)(n0 + row) * ldw1 + 2 * half;
        const long wstep = (long)16 * ldw1;
        for (int k = 0; k < K1; k += 4) {
            v2f av = *(const v2f*)(a + k);
            v2f b0 = *(const v2f*)(w + k);
            v2f b1 = *(const v2f*)(w + wstep + k);
            v2f b2 = *(const v2f*)(w + 2 * wstep + k);
            v2f b3 = *(const v2f*)(w + 3 * wstep + k);
            acc0 = __builtin_amdgcn_wmma_f32_16x16x4_f32(false, av, false, b0, (short)0, acc0, false, false);
            acc1 = __builtin_amdgcn_wmma_f32_16x16x4_f32(false, av, false, b1, (short)0, acc1, false, false);
            acc2 = __builtin_amdgcn_wmma_f32_16x16x4_f32(false, av, false, b2, (short)0, acc2, false, false);
            acc3 = __builtin_amdgcn_wmma_f32_16x16x4_f32(false, av, false, b3, (short)0, acc3, false, false);
        }
    }
    if (A2) {
        const float* a = A2 + (long)(m0 + row) * K2 + 2 * half;
        const float* w = W2 + (long)(n0 + row) * ldw2 + 2 * half;
        const long wstep = (long)16 * ldw2;
        for (int k = 0; k < K2; k += 4) {
            v2f av = *(const v2f*)(a + k);
            v2f b0 = *(const v2f*)(w + k);
            v2f b1 = *(const v2f*)(w + wstep + k);
            v2f b2 = *(const v2f*)(w + 2 * wstep + k);
            v2f b3 = *(const v2f*)(w + 3 * wstep + k);
            acc0 = __builtin_amdgcn_wmma_f32_16x16x4_f32(false, av, false, b0, (short)0, acc0, false, false);
            acc1 = __builtin_amdgcn_wmma_f32_16x16x4_f32(false, av, false, b1, (short)0, acc1, false, false);
            acc2 = __builtin_amdgcn_wmma_f32_16x16x4_f32(false, av, false, b2, (short)0, acc2, false, false);
            acc3 = __builtin_amdgcn_wmma_f32_16x16x4_f32(false, av, false, b3, (short)0, acc3, false, false);
        }
    }

    v8f accs[4] = {acc0, acc1, acc2, acc3};
#pragma unroll
    for (int g = 0; g < 4; ++g) {
        const int n = n0 + (g << 4) + row;        // this lane's output column
        float bias = 0.f;
        if (bias0) bias += bias0[n];
        if (bias1) bias += bias1[n];
#pragma unroll
        for (int j = 0; j < 8; ++j) {             // D layout: M = m0 + 8*half + j
            const int m = m0 + (half << 3) + j;
            float v = accs[g][j] + bias;
            if (act) v = tanhf(v);
            C[(long)m * ldc + col_off + n] = v;
        }
    }
}

// xs[t][b][e]: t==0 -> decoder_input, else label_emb[cur_labels[b][t-1]]
__global__ void build_xs(const float* __restrict__ dec_in,
                         const int* __restrict__ labels,
                         const float* __restrict__ emb,
                         float* __restrict__ xs, int T, int B, int E)
{
    long i = (long)blockIdx.x * blockDim.x + threadIdx.x;
    long total = (long)T * B * E;
    if (i >= total) return;
    int e = (int)(i % E);
    long r = i / E;
    int b = (int)(r % B);
    int t = (int)(r / B);
    float v;
    if (t == 0) v = dec_in[(long)b * E + e];
    else {
        int lbl = labels[(long)b * T + (t - 1)];
        v = emb[(long)lbl * E + e];
    }
    xs[i] = v;
}

// c_t = sig(f)*c + sig(i)*tanh(g);  h_lstm = sig(o)*tanh(c_t)
__global__ void lstm_cell(const float* __restrict__ gates,
                          float* __restrict__ c, float* __restrict__ h_lstm,
                          int B, int H)
{
    int idx = blockIdx.x * blockDim.x + threadIdx.x;
    if (idx >= B * H) return;
    int b = idx / H, j = idx - b * H;
    const float* g = gates + (long)b * 4 * H;
    float ig = g[j], fg = g[H + j], gg = g[2 * H + j], og = g[3 * H + j];
    float si = 1.f / (1.f + expf(-ig));
    float sf = 1.f / (1.f + expf(-fg));
    float so = 1.f / (1.f + expf(-og));
    float cn = sf * c[idx] + si * tanhf(gg);
    c[idx] = cn;
    h_lstm[idx] = so * tanhf(cn);
}

// att[b,s] = sum_h V[h] * tanh(inp[b,h] + ctx[b,s,h]) ; one wave per (b,s)
__global__ __launch_bounds__(256) void attn_scores(
    const float* __restrict__ inp, const float* __restrict__ ctx,
    const float* __restrict__ V, float* __restrict__ att, int B, int S, int H)
{
    int wid = threadIdx.x >> 5, lane = threadIdx.x & 31;
    long id = (long)blockIdx.x * 8 + wid;
    if (id >= (long)B * S) return;
    int b = (int)(id / S);
    const float* crow = ctx + id * H;
    const float* irow = inp + (long)b * H;
    float sum = 0.f;
    for (int h = lane; h < H; h += 32)
        sum += V[h] * tanhf(irow[h] + crow[h]);
    for (int m = 16; m > 0; m >>= 1) sum += __shfl_xor(sum, m, 32);
    if (lane == 0) att[id] = sum;
}

// in-place softmax over each row of length S (one block per row)
__global__ __launch_bounds__(256) void softmax_rows(float* __restrict__ att, int S)
{
    __shared__ float red[256];
    float* row = att + (long)blockIdx.x * S;
    float mx = -INFINITY;
    for (int s = threadIdx.x; s < S; s += 256) mx = fmaxf(mx, row[s]);
    red[threadIdx.x] = mx; __syncthreads();
    for (int off = 128; off > 0; off >>= 1) {
        if (threadIdx.x < off) red[threadIdx.x] = fmaxf(red[threadIdx.x], red[threadIdx.x + off]);
        __syncthreads();
    }
    mx = red[0]; __syncthreads();
    float sum = 0.f;
    for (int s = threadIdx.x; s < S; s += 256) {
        float e = expf(row[s] - mx);
        row[s] = e; sum += e;
    }
    red[threadIdx.x] = sum; __syncthreads();
    for (int off = 128; off > 0; off >>= 1) {
        if (threadIdx.x < off) red[threadIdx.x] += red[threadIdx.x + off];
        __syncthreads();
    }
    float inv = 1.f / red[0];
    for (int s = threadIdx.x; s < S; s += 256) row[s] *= inv;
}

// weighted[b,h] = sum_s alpha[b,s] * ctx[b,s,h] ; block per b, thread per h
__global__ void attn_weighted(const float* __restrict__ alpha,
                              const float* __restrict__ ctx,
                              float* __restrict__ wsum, int S, int H)
{
    int b = blockIdx.x, j = threadIdx.x;
    const float* a = alpha + (long)b * S;
    const float* c = ctx + (long)b * S * H + j;
    float acc = 0.f;
    for (int s = 0; s < S; ++s) acc += a[s] * c[(long)s * H];
    wsum[(long)b * H + j] = acc;
}

// preds[r] = argmax over L of outs[r, :] ; one wave per row
__global__ __launch_bounds__(256) void argmax_rows(
    const float* __restrict__ outs, float* __restrict__ preds, int rows, int L)
{
    int wid = threadIdx.x >> 5, lane = threadIdx.x & 31;
    int r = blockIdx.x * 8 + wid;
    if (r >= rows) return;
    const float* row = outs + (long)r * L;
    float best = -INFINITY; int barg = 0;
    for (int l = lane; l < L; l += 32) {
        float v = row[l];
        if (v > best) { best = v; barg = l; }
    }
    for (int m = 16; m > 0; m >>= 1) {
        float ov = __shfl_xor(best, m, 32);
        int   oa = __shfl_xor(barg, m, 32);
        if (ov > best || (ov == best && oa < barg)) { best = ov; barg = oa; }
    }
    if (lane == 0) preds[r] = (float)barg;
}

__global__ void copy_finals(const float* __restrict__ h, const float* __restrict__ c,
                            float* __restrict__ dst, int n)
{
    int i = blockIdx.x * blockDim.x + threadIdx.x;
    if (i < n) dst[i] = h[i];
    else if (i < 2 * n) dst[i] = c[i - n];
}

extern "C" void kernel_launch(void* const* d_in, const int* in_sizes, int n_in,
                              void* d_out, int out_size, void* d_ws, size_t ws_size,
                              hipStream_t stream)
{
    const int B = in_sizes[0];
    const int E = in_sizes[2] / B;
    const int H = in_sizes[3] / B;
    const int S = in_sizes[5] / (B * H);
    const int T = in_sizes[6] / B;
    const int L = in_sizes[21];
    const int H4 = 4 * H;

    const float* dec_in     = (const float*)d_in[2];
    const float* h0         = (const float*)d_in[3];
    const float* c0         = (const float*)d_in[4];
    const float* context    = (const float*)d_in[5];
    const int*   labels     = (const int*)d_in[6];
    const float* label_emb  = (const float*)d_in[8];
    const float* W_ih       = (const float*)d_in[9];
    const float* b_ih       = (const float*)d_in[10];
    const float* W_hh       = (const float*)d_in[11];
    const float* b_hh       = (const float*)d_in[12];
    const float* W_ho       = (const float*)d_in[13];
    const float* b_ho       = (const float*)d_in[14];
    const float* W_attn_in  = (const float*)d_in[15];
    const float* b_attn_in  = (const float*)d_in[16];
    const float* W_attn_ctx = (const float*)d_in[17];
    const float* b_attn_ctx = (const float*)d_in[18];
    const float* Vp         = (const float*)d_in[19];
    const float* W_out      = (const float*)d_in[20];
    const float* b_out      = (const float*)d_in[21];

    float* outp = (float*)d_out;                      // outputs [B,T,L]
    float* preds = outp + (long)B * T * L;            // preds   [B,T]
    float* hf_dst = preds + (long)B * T;              // h_f,c_f [B,H] each

    // workspace layout (floats)
    float* w = (float*)d_ws;
    float* ctx   = w;                w += (long)B * S * H;
    float* xs    = w;                w += (long)T * B * E;
    float* gates = w;                w += (long)B * H4;
    float* hbuf  = w;                w += (long)B * H;   // carry hidden (hidden_t)
    float* cbuf  = w;                w += (long)B * H;   // carry cell
    float* hlstm = w;                w += (long)B * H;
    float* inp   = w;                w += (long)B * H;
    float* att   = w;                w += (long)B * S;
    float* wsum  = w;                w += (long)B * H;

    // init carries
    (void)hipMemcpyAsync(hbuf, h0, (size_t)B * H * sizeof(float),
                         hipMemcpyDeviceToDevice, stream);
    (void)hipMemcpyAsync(cbuf, c0, (size_t)B * H * sizeof(float),
                         hipMemcpyDeviceToDevice, stream);

    // xs[T,B,E]
    {
        long total = (long)T * B * E;
        build_xs<<<(int)((total + 255) / 256), 256, 0, stream>>>(
            dec_in, labels, label_emb, xs, T, B, E);
    }

    // ctx = context @ W_attn_ctx^T + b_attn_ctx  : [(B*S) x H] = [(B*S) x H][H x H]
    {
        int slabs = ((B * S) >> 4) * (H >> 6);
        wmma_gemm3<<<slabs / 8, 256, 0, stream>>>(
            context, W_attn_ctx, H, H,
            nullptr, nullptr, 0, 0,
            nullptr, nullptr, 0, 0,
            b_attn_ctx, nullptr,
            ctx, B * S, H, H, 0, 0);
    }

    for (int t = 0; t < T; ++t) {
        const float* x_t = xs + (long)t * B * E;

        // gates = x @ W_ih^T + h @ W_hh^T + b_ih + b_hh : [B x 4H]
        {
            int slabs = (B >> 4) * (H4 >> 6);
            wmma_gemm3<<<slabs / 8, 256, 0, stream>>>(
                x_t, W_ih, E, E,
                hbuf, W_hh, H, H,
                nullptr, nullptr, 0, 0,
                b_ih, b_hh,
                gates, B, H4, H4, 0, 0);
        }
        // LSTM cell pointwise
        lstm_cell<<<(B * H + 255) / 256, 256, 0, stream>>>(gates, cbuf, hlstm, B, H);

        // inp = h_lstm @ W_attn_in^T + b_attn_in : [B x H]
        {
            int slabs = (B >> 4) * (H >> 6);
            wmma_gemm3<<<slabs / 8, 256, 0, stream>>>(
                hlstm, W_attn_in, H, H,
                nullptr, nullptr, 0, 0,
                nullptr, nullptr, 0, 0,
                b_attn_in, nullptr,
                inp, B, H, H, 0, 0);
        }
        // attention scores, softmax, weighted sum
        {
            long waves = (long)B * S;
            attn_scores<<<(int)((waves + 7) / 8), 256, 0, stream>>>(
                inp, ctx, Vp, att, B, S, H);
        }
        softmax_rows<<<B, 256, 0, stream>>>(att, S);
        attn_weighted<<<B, H, 0, stream>>>(att, ctx, wsum, S, H);

        // hidden_t = tanh([weighted, h_lstm] @ W_ho^T + b_ho) -> becomes carry h
        {
            int slabs = (B >> 4) * (H >> 6);
            wmma_gemm3<<<slabs / 8, 256, 0, stream>>>(
                wsum, W_ho, H, 2 * H,
                hlstm, W_ho + H, H, 2 * H,
                nullptr, nullptr, 0, 0,
                b_ho, nullptr,
                hbuf, B, H, H, 0, 1);
        }
        // out = [hidden_t, weighted, x] @ W_out^T + b_out -> outputs[:, t, :]
        {
            int ldw = 2 * H + E;
            int slabs = (B >> 4) * (L >> 6);
            wmma_gemm3<<<slabs / 8, 256, 0, stream>>>(
                hbuf, W_out, H, ldw,
                wsum, W_out + H, H, ldw,
                x_t, W_out + 2 * H, E, ldw,
                b_out, nullptr,
                outp, B, L, T * L, t * L, 0);
        }
    }

    // preds = argmax(outputs, -1); copy final h, c
    {
        int rows = B * T;
        argmax_rows<<<(rows + 7) / 8, 256, 0, stream>>>(outp, preds, rows, L);
        copy_finals<<<(2 * B * H + 255) / 256, 256, 0, stream>>>(hbuf, cbuf, hf_dst, B * H);
    }
}